// GraphCNN_54889682042891
// MI455X (gfx1250) — compile-verified
//
#include <hip/hip_runtime.h>
#include <hip/hip_bf16.h>

typedef __attribute__((ext_vector_type(2))) float    v2f;
typedef __attribute__((ext_vector_type(8))) float    v8f;
typedef __attribute__((ext_vector_type(4))) unsigned v4u;
typedef __attribute__((ext_vector_type(8))) int      v8i;
typedef __attribute__((ext_vector_type(4))) int      v4i;

#define N_FEAT   128
#define HIDDEN   128
#define HID2     64
#define N_CLS    10
#define MTILES   5          // 16-row tiles per block -> 80 rows/block, 50000/80 = 625 blocks

#if __has_builtin(__builtin_amdgcn_tensor_load_to_lds) && __has_builtin(__builtin_amdgcn_s_wait_tensorcnt)
#define USE_TDM 1
#else
#define USE_TDM 0
#endif

// ---------------------------------------------------------------------------
// C[M,N] = A[M,K] * B[K,N]  (row-major, fp32) via V_WMMA_F32_16X16X4_F32.
// Whole B (K*N fp32) is staged to LDS once per block with the Tensor Data
// Mover (D# built per ISA ch.8), then each wave computes MTILES 16x16 tiles
// of C with a K-loop of 16x16x4 WMMAs, reading B fragments from LDS.
// blockDim.x = 32 * (N/16); gridDim.x = M / (16*MTILES); shmem = K*N*4.
// Fragment layouts (ISA 7.12.2, 32-bit):
//   A 16x4 : lanes 0-15 rows, VGPR0/1 = K0,K1 ; lanes 16-31 = K2,K3
//   B 4x16 : lanes index N ; VGPR r, lane-half h -> K = r + 2h
//   C 16x16: VGPR r, lane-half h -> M = r + 8h, N = lane&15
// ---------------------------------------------------------------------------
__global__ void gemm_f32_wmma(const float* __restrict__ A,
                              const float* __restrict__ B,
                              float* __restrict__ C,
                              int M, int N, int K) {
    extern __shared__ float Bs[];
    const int lane = threadIdx.x & 31;
    const int wave = threadIdx.x >> 5;
    const int half = lane >> 4;
    const int l15  = lane & 15;
    const int ncol = wave * 16 + l15;

#if USE_TDM
    if (wave == 0) {
        // ---- Tensor DMA descriptor (D#): 2-D tensor, tile == whole K x N ----
        unsigned lds_base = (unsigned)(unsigned long long)
                            (__attribute__((address_space(3))) float*)Bs;
        unsigned long long ga = (unsigned long long)(const void*)B;
        v4u g0;
        g0.x = 1u;                                        // count=1 (valid), user mode
        g0.y = lds_base;                                  // lds_addr [63:32]
        g0.z = (unsigned)ga;                              // global_addr [95:64]
        g0.w = (unsigned)((ga >> 32) & 0x01FFFFFFull)     // global_addr [120:96]
               | (2u << 30);                              // type=2 ("image")
        v8i g1;
        g1[0] = 0x00020000;                               // data_size=2 (4B), mask=0
        g1[1] = (N & 0xFFFF) << 16;                       // tensor_dim0[15:0]
        g1[2] = (K & 0xFFFF) << 16;                       // dim0 hi=0 | tensor_dim1[15:0]
        g1[3] = (N & 0xFFFF) << 16;                       // dim1 hi=0 | tile_dim0=N
        g1[4] = (K & 0xFFFF);                             // tile_dim1=K, tile_dim2=0
        g1[5] = N;                                        // tensor_dim0_stride lo
        g1[6] = 0;                                        // stride hi / dim1_stride lo
        g1[7] = 0;
        v4i gz4 = {0, 0, 0, 0};                           // groups 2/3 unused (2-D)
        v8i gz8 = {0, 0, 0, 0, 0, 0, 0, 0};               // extra group (clang-23 form)
        __builtin_amdgcn_tensor_load_to_lds(g0, g1, gz4, gz4, gz8, 0);
        __builtin_amdgcn_s_wait_tensorcnt(0);
    }
    __syncthreads();
#else
    for (int c = threadIdx.x; c < (K * N) >> 2; c += blockDim.x)
        ((float4*)Bs)[c] = ((const float4*)B)[c];
    __syncthreads();
#endif

    for (int mt = 0; mt < MTILES; ++mt) {
        const int m0 = (blockIdx.x * MTILES + mt) * 16;
        const long long arow = (long long)(m0 + l15) * K;
        v8f acc = {};
        for (int k = 0; k < K; k += 4) {
            v2f a, b;
            const float* ap = A + arow + k + 2 * half;
            a.x = ap[0];
            a.y = ap[1];
            b.x = Bs[(k + 2 * half + 0) * N + ncol];
            b.y = Bs[(k + 2 * half + 1) * N + ncol];
            acc = __builtin_amdgcn_wmma_f32_16x16x4_f32(
                /*neg_a=*/false, a, /*neg_b=*/false, b,
                /*c_mod=*/(short)0, acc, /*reuse_a=*/false, /*reuse_b=*/false);
        }
#pragma unroll
        for (int r = 0; r < 8; ++r) {
            C[(long long)(m0 + r + 8 * half) * N + ncol] = acc[r];
        }
    }
}

// ---------------------------------------------------------------------------
// Degree / normalization
// ---------------------------------------------------------------------------
__global__ void init_deg(float* __restrict__ deg, int n) {
    int i = blockIdx.x * blockDim.x + threadIdx.x;
    if (i < n) deg[i] = 1.0f;                 // self-loop contribution
}

__global__ void add_deg(const int* __restrict__ col, float* __restrict__ deg, int E) {
    int e = blockIdx.x * blockDim.x + threadIdx.x;
    if (e < E) atomicAdd(&deg[col[e]], 1.0f);
}

__global__ void finish_dinv(float* __restrict__ deg, int n) {
    int i = blockIdx.x * blockDim.x + threadIdx.x;
    if (i < n) deg[i] = rsqrtf(deg[i]);       // deg >= 1 always (self-loop)
}

__global__ void zerof(float* __restrict__ p, long long n) {
    long long i = (long long)blockIdx.x * blockDim.x + threadIdx.x;
    if (i < n) p[i] = 0.0f;
}

// ---------------------------------------------------------------------------
// Edge scatter: AGG[col[e]][:] += T[row[e]][:] * dinv[row]*dinv[col]
// One thread handles one float4 chunk of one edge.
// ---------------------------------------------------------------------------
__global__ void scatter_edges(const float* __restrict__ T,
                              float* __restrict__ AGG,
                              const int* __restrict__ row,
                              const int* __restrict__ col,
                              const float* __restrict__ dinv,
                              int E, int F) {
    const int chunks = F >> 2;
    long long tid = (long long)blockIdx.x * blockDim.x + threadIdx.x;
    long long total = (long long)E * chunks;
    if (tid >= total) return;
    int e = (int)(tid / chunks);
    int c = (int)(tid % chunks);
    int r  = row[e];
    int cl = col[e];
    float nrm = dinv[r] * dinv[cl];
    float4 v = *((const float4*)(T + (long long)r * F) + c);
    float* dst = AGG + (long long)cl * F + (c << 2);
    atomicAdd(dst + 0, v.x * nrm);
    atomicAdd(dst + 1, v.y * nrm);
    atomicAdd(dst + 2, v.z * nrm);
    atomicAdd(dst + 3, v.w * nrm);
}

// H = relu(AGG + T * dinv^2 + b)   (self-loop term folded in here, no atomics)
__global__ void bias_relu_selfloop(const float* __restrict__ AGG,
                                   const float* __restrict__ T,
                                   const float* __restrict__ dinv,
                                   const float* __restrict__ b,
                                   float* __restrict__ H,
                                   int n, int F) {
    long long idx = (long long)blockIdx.x * blockDim.x + threadIdx.x;
    if (idx >= (long long)n * F) return;
    int node = (int)(idx / F);
    int f    = (int)(idx % F);
    float d = dinv[node];
    float v = AGG[idx] + T[idx] * d * d + b[f];
    H[idx] = fmaxf(v, 0.0f);
}

// ---------------------------------------------------------------------------
// Global mean pool
// ---------------------------------------------------------------------------
__global__ void pool_feat(const float* __restrict__ H,
                          const int* __restrict__ batch,
                          float* __restrict__ sums,
                          int n, int F) {
    long long idx = (long long)blockIdx.x * blockDim.x + threadIdx.x;
    if (idx >= (long long)n * F) return;
    int node = (int)(idx / F);
    int f    = (int)(idx % F);
    atomicAdd(&sums[batch[node] * F + f], H[idx]);
}

__global__ void pool_cnt(const int* __restrict__ batch,
                         float* __restrict__ cnt, int n) {
    int i = blockIdx.x * blockDim.x + threadIdx.x;
    if (i < n) atomicAdd(&cnt[batch[i]], 1.0f);
}

// ---------------------------------------------------------------------------
// MLP head: pooled[g] = sums[g]/max(cnt,1); relu(pooled@Wf1+bf1) @ Wf2 + bf2
// One block per graph.
// ---------------------------------------------------------------------------
__global__ void mlp_head(const float* __restrict__ sums,
                         const float* __restrict__ cnt,
                         const float* __restrict__ Wf1,
                         const float* __restrict__ bf1,
                         const float* __restrict__ Wf2,
                         const float* __restrict__ bf2,
                         float* __restrict__ out) {
    __shared__ float pooled[HID2];
    __shared__ float hfc[32];
    const int g = blockIdx.x;
    const int t = threadIdx.x;
    const float c = fmaxf(cnt[g], 1.0f);
    if (t < HID2) pooled[t] = sums[g * HID2 + t] / c;
    __syncthreads();
    if (t < 32) {
        float acc = bf1[t];
        for (int k = 0; k < HID2; ++k) acc += pooled[k] * Wf1[k * 32 + t];
        hfc[t] = fmaxf(acc, 0.0f);
    }
    __syncthreads();
    if (t < N_CLS) {
        float acc = bf2[t];
        for (int k = 0; k < 32; ++k) acc += hfc[k] * Wf2[k * N_CLS + t];
        out[g * N_CLS + t] = acc;
    }
}

// ---------------------------------------------------------------------------
// Launch
// ---------------------------------------------------------------------------
extern "C" void kernel_launch(void* const* d_in, const int* in_sizes, int n_in,
                              void* d_out, int out_size, void* d_ws, size_t ws_size,
                              hipStream_t stream) {
    (void)n_in; (void)ws_size;

    const float* x     = (const float*)d_in[0];
    const int*   ei    = (const int*)  d_in[1];
    const int*   batch = (const int*)  d_in[2];
    const float* W1    = (const float*)d_in[3];
    const float* b1    = (const float*)d_in[4];
    const float* W2    = (const float*)d_in[5];
    const float* b2    = (const float*)d_in[6];
    const float* W3    = (const float*)d_in[7];
    const float* b3    = (const float*)d_in[8];
    const float* Wf1   = (const float*)d_in[9];
    const float* bf1   = (const float*)d_in[10];
    const float* Wf2   = (const float*)d_in[11];
    const float* bf2   = (const float*)d_in[12];
    float* out = (float*)d_out;

    const int NN = in_sizes[0] / N_FEAT;      // 50000
    const int E  = in_sizes[1] / 2;           // 800000
    const int G  = out_size / N_CLS;          // 100
    const int* row = ei;
    const int* col = ei + E;

    // Workspace layout (floats)
    float* ws   = (float*)d_ws;
    float* T    = ws;                                   // [NN,128]
    float* AG   = T  + (long long)NN * HIDDEN;          // [NN,128]
    float* H    = AG + (long long)NN * HIDDEN;          // [NN,128]
    float* dinv = H  + (long long)NN * HIDDEN;          // [NN]
    float* sums = dinv + NN;                            // [G,64]
    float* cnt  = sums + (long long)G * HID2;           // [G]

    const int BT = 256;
    auto cdiv = [](long long a, long long b) { return (int)((a + b - 1) / b); };

    const int gemm_grid = NN / (16 * MTILES);           // 625

    // --- normalization: dinv = rsqrt(1 + in-degree) ---
    init_deg   <<<cdiv(NN, BT), BT, 0, stream>>>(dinv, NN);
    add_deg    <<<cdiv(E,  BT), BT, 0, stream>>>(col, dinv, E);
    finish_dinv<<<cdiv(NN, BT), BT, 0, stream>>>(dinv, NN);

    // layer 1: x[NN,128] -> H[NN,128]
    gemm_f32_wmma<<<gemm_grid, 32 * (HIDDEN / 16), N_FEAT * HIDDEN * 4, stream>>>(x, W1, T, NN, HIDDEN, N_FEAT);
    zerof<<<cdiv((long long)NN * HIDDEN, BT), BT, 0, stream>>>(AG, (long long)NN * HIDDEN);
    scatter_edges<<<cdiv((long long)E * (HIDDEN / 4), BT), BT, 0, stream>>>(T, AG, row, col, dinv, E, HIDDEN);
    bias_relu_selfloop<<<cdiv((long long)NN * HIDDEN, BT), BT, 0, stream>>>(AG, T, dinv, b1, H, NN, HIDDEN);

    // layer 2: H[NN,128] -> H[NN,128]
    gemm_f32_wmma<<<gemm_grid, 32 * (HIDDEN / 16), HIDDEN * HIDDEN * 4, stream>>>(H, W2, T, NN, HIDDEN, HIDDEN);
    zerof<<<cdiv((long long)NN * HIDDEN, BT), BT, 0, stream>>>(AG, (long long)NN * HIDDEN);
    scatter_edges<<<cdiv((long long)E * (HIDDEN / 4), BT), BT, 0, stream>>>(T, AG, row, col, dinv, E, HIDDEN);
    bias_relu_selfloop<<<cdiv((long long)NN * HIDDEN, BT), BT, 0, stream>>>(AG, T, dinv, b2, H, NN, HIDDEN);

    // layer 3: H[NN,128] -> H[NN,64]
    gemm_f32_wmma<<<gemm_grid, 32 * (HID2 / 16), HIDDEN * HID2 * 4, stream>>>(H, W3, T, NN, HID2, HIDDEN);
    zerof<<<cdiv((long long)NN * HID2, BT), BT, 0, stream>>>(AG, (long long)NN * HID2);
    scatter_edges<<<cdiv((long long)E * (HID2 / 4), BT), BT, 0, stream>>>(T, AG, row, col, dinv, E, HID2);
    bias_relu_selfloop<<<cdiv((long long)NN * HID2, BT), BT, 0, stream>>>(AG, T, dinv, b3, H, NN, HID2);

    // --- global mean pool + MLP head ---
    zerof<<<cdiv((long long)G * HID2 + G, BT), BT, 0, stream>>>(sums, (long long)G * HID2 + G);
    pool_feat<<<cdiv((long long)NN * HID2, BT), BT, 0, stream>>>(H, batch, sums, NN, HID2);
    pool_cnt <<<cdiv(NN, BT), BT, 0, stream>>>(batch, cnt, NN);
    mlp_head <<<G, 64, 0, stream>>>(sums, cnt, Wf1, bf1, Wf2, bf2, out);
}